// Ortho_fold_branch_34291018891882
// MI455X (gfx1250) — compile-verified
//
#include <hip/hip_runtime.h>

typedef __attribute__((ext_vector_type(2))) float v2f;
typedef __attribute__((ext_vector_type(8))) float v8f;

#define IMG_H 256
#define IMG_W 256
#define NCH   64          // patch dim = 8*8
#define NIMG  16          // b*c
#define LDSW  136         // x-tile LDS row stride (135 cols needed, pad to 136)
#define ATW   65          // A^T LDS row stride (64 + 1 pad -> conflict-free A-frag loads)

// ---------------------------------------------------------------------------
// Kernel 1: compose A = H0*H1*...*H63 from Householder vectors v (64x64),
// write A^T (At[m*64+k] = A[k][m]) to workspace. One block of 64 threads;
// thread t owns row t of A in registers (64 VGPRs).
// ---------------------------------------------------------------------------
__global__ void __launch_bounds__(64)
ortho_compose_householder(const float* __restrict__ v, float* __restrict__ At)
{
    __shared__ float lv[NCH * NCH];
    const int t = threadIdx.x;                 // 0..63
    for (int j = t; j < NCH * NCH; j += 64) lv[j] = v[j];
    __syncthreads();

    float Ar[NCH];
#pragma unroll
    for (int j = 0; j < NCH; ++j) Ar[j] = (j == t) ? 1.0f : 0.0f;

    for (int i = 0; i < NCH; ++i) {
        const float* vi = &lv[i * NCH];
        float s = 0.0f, d = 0.0f;
#pragma unroll
        for (int j = 0; j < NCH; ++j) {
            s += vi[j] * vi[j];                // ||v_i||^2 (redundant per thread, tiny)
            d += Ar[j] * vi[j];                // (A v_i)[t]
        }
        const float c = 2.0f * d / s;
#pragma unroll
        for (int j = 0; j < NCH; ++j) Ar[j] -= c * vi[j];   // A <- A * H_i, row t
    }
    // Ar[m] = A[t][m]  ->  At[m][k=t]
#pragma unroll
    for (int m = 0; m < NCH; ++m) At[m * NCH + t] = Ar[m];
}

__device__ __forceinline__ int reflect_idx(int p, int n) {
    // jnp.pad 'reflect' (mirror, edge not repeated); pad < n so one fold suffices
    if (p < 0)      return -p;
    if (p > n - 1)  return 2 * (n - 1) - p;
    return p;
}

// ---------------------------------------------------------------------------
// Kernel 2: im2col (reflect-pad, 8x8 patches) fused with GEMM against A via
// V_WMMA_F32_16X16X4_F32. Block = 256 thr (8 waves), handles one image row h,
// 128 consecutive columns. Wave w owns 16 columns x all 64 output channels.
// grid = (2 wchunks, 256 rows, 16 images)
// ---------------------------------------------------------------------------
__global__ void __launch_bounds__(256)
ortho_unfold_wmma(const float* __restrict__ x, const float* __restrict__ At,
                  float* __restrict__ out)
{
    __shared__ float sAt[NCH * ATW];       // A^T, row stride 65 (bank-conflict free)
    __shared__ float sX[8 * LDSW];         // reflected input band: 8 rows x 135 cols

    const int wchunk = blockIdx.x;         // 0..1
    const int h      = blockIdx.y;         // 0..255
    const int b      = blockIdx.z;         // 0..15
    const int tid    = threadIdx.x;
    const int wbase  = wchunk * 128;

    // Stage A^T with +1 padded stride
    for (int j = tid; j < NCH * NCH; j += 256) {
        const int m = j >> 6, k = j & 63;
        sAt[m * ATW + k] = At[j];
    }
    // Stage input band with reflection applied (padded rows h..h+7, cols wbase-3..wbase+131+..)
    const float* xb = x + (size_t)b * (IMG_H * IMG_W);
    for (int idx = tid; idx < 8 * 135; idx += 256) {
        const int ry = idx / 135, cx = idx - ry * 135;
        const int pr = reflect_idx(h - 3 + ry, IMG_H);
        const int pc = reflect_idx(wbase - 3 + cx, IMG_W);
        sX[ry * LDSW + cx] = xb[pr * IMG_W + pc];
    }
    __syncthreads();

    const int lane = tid & 31;
    const int wave = tid >> 5;             // 0..7
    const int half = lane >> 4;            // 0/1 -> K offset per WMMA layout
    const int n    = lane & 15;            // WMMA-N lane (spatial) == WMMA-M lane (channel)
    const int lw   = wave * 16 + n;        // local spatial column in sX

    v8f acc[4] = {};                       // 4 M-tiles x 16x16 f32 accumulators

#pragma unroll
    for (int kk = 0; kk < 16; ++kk) {      // K = 64 in chunks of 4
        const int k0 = kk * 4 + half * 2;  // lane's first K (VGPR0); VGPR1 = k0+1
        const int ky = k0 >> 3, kx = k0 & 7;   // k -> (patch row, patch col); same ky for k0,k0+1
        v2f bf;                            // B-frag: cols[k, l]
        bf.x = sX[ky * LDSW + lw + kx];
        bf.y = sX[ky * LDSW + lw + kx + 1];
#pragma unroll
        for (int mt = 0; mt < 4; ++mt) {
            v2f af;                        // A-frag: A^T[m, k]
            af.x = sAt[(mt * 16 + n) * ATW + k0];
            af.y = sAt[(mt * 16 + n) * ATW + k0 + 1];
            acc[mt] = __builtin_amdgcn_wmma_f32_16x16x4_f32(
                false, af, false, bf, (short)0, acc[mt], false, false);
        }
    }

    // Store: D vgpr r -> channel m = mt*16 + r + 8*half, lanes 0..15 = consecutive l.
    // Output never re-read and >> L2: stream it with non-temporal stores.
    float* ob = out + (size_t)b * NCH * (IMG_H * IMG_W)
                    + (size_t)h * IMG_W + wbase + wave * 16 + n;
#pragma unroll
    for (int mt = 0; mt < 4; ++mt) {
#pragma unroll
        for (int r = 0; r < 8; ++r) {
            const int m = mt * 16 + r + 8 * half;
            __builtin_nontemporal_store(acc[mt][r], ob + (size_t)m * (IMG_H * IMG_W));
        }
    }
}

extern "C" void kernel_launch(void* const* d_in, const int* in_sizes, int n_in,
                              void* d_out, int out_size, void* d_ws, size_t ws_size,
                              hipStream_t stream)
{
    const float* x = (const float*)d_in[0];   // (2,8,256,256) fp32 -> 16 images
    const float* v = (const float*)d_in[1];   // (64,64) fp32
    float* out = (float*)d_out;               // (16,64,256,256) fp32
    float* At  = (float*)d_ws;                // 64*64 fp32 = 16 KB scratch

    ortho_compose_householder<<<1, 64, 0, stream>>>(v, At);

    dim3 grid(2, IMG_H, NIMG);
    ortho_unfold_wmma<<<grid, 256, 0, stream>>>(x, At, out);
}